// S2SMLayer_21818433864133
// MI455X (gfx1250) — compile-verified
//
#include <hip/hip_runtime.h>
#include <hip/hip_bf16.h>

typedef float v2f __attribute__((ext_vector_type(2)));
typedef float v8f __attribute__((ext_vector_type(8)));

#define B_  16
#define L_  2048
#define D_  128
#define F_  100
#define KS_ 3
#define HMAX 8          // max recorded top-k hits per row/col (k=3 + tie slack)
#define CONV_TL 8       // conv positions per block

// ---------------------------------------------------------------------------
// Kernel 1: W[b] = context[b] (LxD) * main[b]^T (DxL) via V_WMMA_F32_16X16X4_F32.
// One wave computes a 32x32 tile (4 accumulators): 4 fragment loads -> 4 WMMAs
// per K-step, doubling matrix-op density per byte vs a single 16x16 tile.
// A-frag (16x4 f32): M = lane&15, VGPR0/1 hold K = khalf, khalf+1, khalf=(lane>=16)?2:0.
// D-frag (16x16 f32): N = lane&15, VGPR v -> M = v + (lane>=16 ? 8 : 0).
// ---------------------------------------------------------------------------
__global__ __launch_bounds__(128) void s2sm_gemm_wmma(
    const float* __restrict__ ctx, const float* __restrict__ mn,
    float* __restrict__ W)
{
    const int lane = threadIdx.x & 31;
    const int wv   = threadIdx.x >> 5;
    const int b    = blockIdx.z;
    const int col0 = blockIdx.x * 32;
    const int row0 = (blockIdx.y * 4 + wv) * 32;

    const int mi    = lane & 15;
    const int khalf = (lane >> 4) << 1;   // 0 or 2

    const float* a0p = ctx + ((size_t)b * L_ + row0 + mi) * D_ + khalf;
    const float* a1p = a0p + 16 * D_;
    const float* b0p = mn  + ((size_t)b * L_ + col0 + mi) * D_ + khalf;
    const float* b1p = b0p + 16 * D_;

    __builtin_prefetch(a0p, 0, 3);
    __builtin_prefetch(b0p, 0, 3);

    v8f acc00 = {0.f,0.f,0.f,0.f,0.f,0.f,0.f,0.f};
    v8f acc01 = acc00, acc10 = acc00, acc11 = acc00;

    #pragma unroll 4
    for (int k0 = 0; k0 < D_; k0 += 4) {
        v2f a0 = *(const v2f*)(a0p + k0);
        v2f a1 = *(const v2f*)(a1p + k0);
        v2f b0 = *(const v2f*)(b0p + k0);
        v2f b1 = *(const v2f*)(b1p + k0);
        acc00 = __builtin_amdgcn_wmma_f32_16x16x4_f32(false, a0, false, b0, (short)0, acc00, false, false);
        acc01 = __builtin_amdgcn_wmma_f32_16x16x4_f32(false, a0, false, b1, (short)0, acc01, false, false);
        acc10 = __builtin_amdgcn_wmma_f32_16x16x4_f32(false, a1, false, b0, (short)0, acc10, false, false);
        acc11 = __builtin_amdgcn_wmma_f32_16x16x4_f32(false, a1, false, b1, (short)0, acc11, false, false);
    }

    const int n  = lane & 15;
    const int mo = (lane >> 4) * 8;
    float* base  = W + ((size_t)b * L_ + row0 + mo) * L_ + col0 + n;
    float* base2 = base + (size_t)16 * L_;
    #pragma unroll
    for (int v = 0; v < 8; ++v) {
        base [(size_t)v * L_]      = acc00[v];
        base [(size_t)v * L_ + 16] = acc01[v];
        base2[(size_t)v * L_]      = acc10[v];
        base2[(size_t)v * L_ + 16] = acc11[v];
    }
}

// ---------------------------------------------------------------------------
// top-3 helpers
// ---------------------------------------------------------------------------
__device__ __forceinline__ void push3(float v, float& t0, float& t1, float& t2) {
    if (v > t2) {
        if (v > t1) { t2 = t1; if (v > t0) { t1 = t0; t0 = v; } else t1 = v; }
        else t2 = v;
    }
}

__device__ __forceinline__ void merge3(float& a0, float& a1, float& a2,
                                       float b0, float b1, float b2) {
    float v[6] = {a0, a1, a2, b0, b1, b2};
    #pragma unroll
    for (int i = 0; i < 3; ++i) {
        int mx = i;
        #pragma unroll
        for (int j = i + 1; j < 6; ++j) if (v[j] > v[mx]) mx = j;
        float t = v[i]; v[i] = v[mx]; v[mx] = t;
    }
    a0 = v[0]; a1 = v[1]; a2 = v[2];
}

// ---------------------------------------------------------------------------
// Kernel 2: per-row stats + hit list. One 256-thread block per (b,row).
// Single global pass: each thread keeps its 8 values in registers and replays
// them for the kept-sum and top-k hit recording (saves a full 256MB W pass and
// removes the scatter kernel's W scan entirely).
// ---------------------------------------------------------------------------
__global__ __launch_bounds__(256) void s2sm_row_stats(
    const float* __restrict__ W, float* __restrict__ kth,
    float* __restrict__ inv, float* __restrict__ stdv,
    int* __restrict__ hitCnt, int* __restrict__ hitIdx, float* __restrict__ hitVal)
{
    const int b = blockIdx.y, r = blockIdx.x, t = threadIdx.x;
    const size_t o = (size_t)b * L_ + r;
    const float* row = W + o * L_;

    float vals[8];
    float t0 = -3.4e38f, t1 = -3.4e38f, t2 = -3.4e38f;
    float s = 0.f, s2 = 0.f;
    #pragma unroll
    for (int j = 0; j < 8; ++j) {
        float v = row[t + j * 256];
        vals[j] = v;
        s += v; s2 += v * v;
        push3(v, t0, t1, t2);
    }

    __shared__ float sh0[256], sh1[256], sh2[256], ss[256], ss2[256];
    __shared__ int hcnt;
    sh0[t] = t0; sh1[t] = t1; sh2[t] = t2; ss[t] = s; ss2[t] = s2;
    __syncthreads();
    for (int off = 128; off > 0; off >>= 1) {
        if (t < off) {
            float a0 = sh0[t], a1 = sh1[t], a2 = sh2[t];
            merge3(a0, a1, a2, sh0[t + off], sh1[t + off], sh2[t + off]);
            sh0[t] = a0; sh1[t] = a1; sh2[t] = a2;
            ss[t]  += ss[t + off];
            ss2[t] += ss2[t + off];
        }
        __syncthreads();
    }
    const float k3    = sh2[0];
    const float totS  = ss[0];
    const float totS2 = ss2[0];
    __syncthreads();
    if (t == 0) hcnt = 0;
    __syncthreads();

    // register replay: kept-sum (exact under ties) + hit recording
    float ks = 0.f;
    #pragma unroll
    for (int j = 0; j < 8; ++j) {
        float v = vals[j];
        if (v >= k3) {
            ks += v;
            int p = atomicAdd(&hcnt, 1);
            if (p < HMAX) { hitIdx[o * HMAX + p] = t + j * 256; hitVal[o * HMAX + p] = v; }
        }
    }
    ss[t] = ks;
    __syncthreads();
    for (int off = 128; off > 0; off >>= 1) {
        if (t < off) ss[t] += ss[t + off];
        __syncthreads();
    }
    if (t == 0) {
        kth[o] = k3;
        inv[o] = 1.0f / ss[0];
        float mean = totS * (1.0f / L_);
        float var  = fmaxf(totS2 * (1.0f / L_) - mean * mean, 0.f);
        stdv[o] = sqrtf(var);
        hitCnt[o] = hcnt < HMAX ? hcnt : HMAX;
    }
}

// ---------------------------------------------------------------------------
// Kernel 3: per-column stats + hit list (thread-per-column streaming, fully
// coalesced). Second pass computes the exact kept-sum and records hit rows,
// removing the gather kernel's W scan.
// ---------------------------------------------------------------------------
__global__ __launch_bounds__(256) void s2sm_col_stats(
    const float* __restrict__ W, float* __restrict__ kth,
    float* __restrict__ inv, float* __restrict__ stdv,
    int* __restrict__ hitCnt, int* __restrict__ hitIdx, float* __restrict__ hitVal)
{
    const int b   = blockIdx.y;
    const int col = blockIdx.x * 256 + threadIdx.x;
    const size_t o = (size_t)b * L_ + col;
    const float* base = W + (size_t)b * L_ * L_ + col;

    float t0 = -3.4e38f, t1 = -3.4e38f, t2 = -3.4e38f;
    float s = 0.f, s2 = 0.f;
    for (int r = 0; r < L_; ++r) {
        float v = base[(size_t)r * L_];
        s += v; s2 += v * v;
        push3(v, t0, t1, t2);
    }
    float ks = 0.f;
    int cnt = 0;
    for (int r = 0; r < L_; ++r) {
        float v = base[(size_t)r * L_];
        if (v >= t2) {
            ks += v;
            if (cnt < HMAX) { hitIdx[o * HMAX + cnt] = r; hitVal[o * HMAX + cnt] = v; ++cnt; }
        }
    }
    kth[o] = t2;
    inv[o] = 1.0f / ks;
    float mean = s * (1.0f / L_);
    float var  = fmaxf(s2 * (1.0f / L_) - mean * mean, 0.f);
    stdv[o] = sqrtf(var);
    hitCnt[o] = cnt;
}

// ---------------------------------------------------------------------------
// Kernel 4: zero-fill
// ---------------------------------------------------------------------------
__global__ void s2sm_zero(float* __restrict__ p, long long n) {
    for (long long i = (long long)blockIdx.x * blockDim.x + threadIdx.x; i < n;
         i += (long long)gridDim.x * blockDim.x)
        p[i] = 0.f;
}

// ---------------------------------------------------------------------------
// Kernel 5: att_merge_c scatter from precomputed row hit lists (no W read):
//   attC[b,m,:] += wk_c[i,m] * context[b,i,:]
// ---------------------------------------------------------------------------
__global__ __launch_bounds__(128) void s2sm_scatter_c(
    const float* __restrict__ ctx, const float* __restrict__ rowInv,
    const int* __restrict__ hitCnt, const int* __restrict__ hitIdx,
    const float* __restrict__ hitVal, float* __restrict__ attC)
{
    const int b = blockIdx.y, i = blockIdx.x, t = threadIdx.x;
    const size_t ro = (size_t)b * L_ + i;
    const int   cnt = hitCnt[ro];
    const float iv  = rowInv[ro];

    __shared__ int   sIdx[HMAX];
    __shared__ float sCo[HMAX];
    if (t < cnt) { sIdx[t] = hitIdx[ro * HMAX + t]; sCo[t] = hitVal[ro * HMAX + t] * iv; }
    __syncthreads();

    const float x = ctx[ro * D_ + t];
    for (int e = 0; e < cnt; ++e)
        atomicAdd(&attC[((size_t)b * L_ + sIdx[e]) * D_ + t], sCo[e] * x);
}

// ---------------------------------------------------------------------------
// Kernel 6: fused att_merge_m gather (from col hit lists) + outputs_c:
//   outc[b,m,d] = |ctx[b,m,d] - sum_e coef_e * main[b,r_e,d]| * std_c[m]
// ---------------------------------------------------------------------------
__global__ __launch_bounds__(128) void s2sm_gather_m_outc(
    const float* __restrict__ mn, const float* __restrict__ ctx,
    const float* __restrict__ colInv, const int* __restrict__ hitCnt,
    const int* __restrict__ hitIdx, const float* __restrict__ hitVal,
    const float* __restrict__ rowStd, float* __restrict__ outc)
{
    const int b = blockIdx.y, m = blockIdx.x, t = threadIdx.x;
    const size_t co = (size_t)b * L_ + m;
    const int   cnt = hitCnt[co];
    const float iv  = colInv[co];

    __shared__ int   sIdx[HMAX];
    __shared__ float sCo[HMAX];
    if (t < cnt) { sIdx[t] = hitIdx[co * HMAX + t]; sCo[t] = hitVal[co * HMAX + t] * iv; }
    __syncthreads();

    float acc = 0.f;
    for (int e = 0; e < cnt; ++e)
        acc = fmaf(sCo[e], mn[((size_t)b * L_ + sIdx[e]) * D_ + t], acc);
    outc[co * D_ + t] = fabsf(ctx[co * D_ + t] - acc) * rowStd[co];
}

// ---------------------------------------------------------------------------
// Kernel 7: outputs_m = |main - attC| * std_m   (elementwise)
// ---------------------------------------------------------------------------
__global__ void s2sm_outm(
    const float* __restrict__ mn, const float* __restrict__ attC,
    const float* __restrict__ colStd, float* __restrict__ outm)
{
    const long long total = (long long)B_ * L_ * D_;
    for (long long i = (long long)blockIdx.x * blockDim.x + threadIdx.x; i < total;
         i += (long long)gridDim.x * blockDim.x) {
        long long bl = i / D_;   // b*L + m
        outm[i] = fabsf(mn[i] - attC[i]) * colStd[bl];
    }
}

// ---------------------------------------------------------------------------
// Kernel 8: conv1d(ksize=3, valid) + ReLU + max-over-L.
// 8 positions per block: each weight load is amortized over 8 FMAs from LDS
// (cuts the 150KB weight tensor's re-read traffic 8x). Post-ReLU >= 0, so
// int-bit atomicMax == float max (out pre-zeroed).
// ---------------------------------------------------------------------------
__global__ __launch_bounds__(128) void s2sm_conv_pool(
    const float* __restrict__ x, const float* __restrict__ wgt,
    const float* __restrict__ bias, float* __restrict__ out, int outOffset)
{
    const int b  = blockIdx.y;
    const int l0 = blockIdx.x * CONV_TL;
    const int t  = threadIdx.x;
    __shared__ float xs[(CONV_TL + KS_ - 1) * D_];   // 10 rows x 128

    const int rows = min(CONV_TL + KS_ - 1, L_ - l0);
    const float* xb = x + ((size_t)b * L_ + l0) * D_;   // contiguous rows
    for (int j = t; j < rows * D_; j += 128) xs[j] = xb[j];
    __syncthreads();

    if (t < F_) {
        float acc[CONV_TL];
        #pragma unroll
        for (int u = 0; u < CONV_TL; ++u) acc[u] = bias[t];
        for (int j = 0; j < KS_ * D_; ++j) {
            float wv = wgt[(size_t)j * F_ + t];
            #pragma unroll
            for (int u = 0; u < CONV_TL; ++u)
                acc[u] = fmaf(xs[u * D_ + j], wv, acc[u]);
        }
        float mx = 0.f;
        #pragma unroll
        for (int u = 0; u < CONV_TL; ++u)
            if (l0 + u < L_ - KS_ + 1) mx = fmaxf(mx, fmaxf(acc[u], 0.f));
        atomicMax((int*)&out[(size_t)b * (2 * F_) + outOffset + t],
                  __float_as_int(mx));
    }
}

// ---------------------------------------------------------------------------
extern "C" void kernel_launch(void* const* d_in, const int* in_sizes, int n_in,
                              void* d_out, int out_size, void* d_ws, size_t ws_size,
                              hipStream_t stream) {
    const float* ctx    = (const float*)d_in[0];   // [B,L,D]
    const float* mn     = (const float*)d_in[1];   // [B,L,D]
    const float* conv_w = (const float*)d_in[2];   // [3,D,F]
    const float* conv_b = (const float*)d_in[3];   // [F]
    float* out = (float*)d_out;                    // [B, 2F]

    // workspace layout
    const size_t nW   = (size_t)B_ * L_ * L_;
    const size_t nBL  = (size_t)B_ * L_;
    const size_t nBLD = (size_t)B_ * L_ * D_;

    char* p = (char*)d_ws;
    float* W       = (float*)p; p += nW * 4;
    float* rowKth  = (float*)p; p += nBL * 4;
    float* rowInv  = (float*)p; p += nBL * 4;
    float* rowStd  = (float*)p; p += nBL * 4;
    float* colKth  = (float*)p; p += nBL * 4;
    float* colInv  = (float*)p; p += nBL * 4;
    float* colStd  = (float*)p; p += nBL * 4;
    int*   rowHitCnt = (int*)p;   p += nBL * 4;
    int*   rowHitIdx = (int*)p;   p += nBL * HMAX * 4;
    float* rowHitVal = (float*)p; p += nBL * HMAX * 4;
    int*   colHitCnt = (int*)p;   p += nBL * 4;
    int*   colHitIdx = (int*)p;   p += nBL * HMAX * 4;
    float* colHitVal = (float*)p; p += nBL * HMAX * 4;
    float* attC = (float*)p; p += nBLD * 4;
    float* outm = (float*)p; p += nBLD * 4;
    float* outc = (float*)p; p += nBLD * 4;

    // zero the scatter accumulator and the atomic-max output
    s2sm_zero<<<4096, 256, 0, stream>>>(attC, (long long)nBLD);
    s2sm_zero<<<16, 256, 0, stream>>>(out, (long long)B_ * 2 * F_);

    // 1) W = ctx @ mn^T (f32 WMMA, 32x32 tile per wave)
    s2sm_gemm_wmma<<<dim3(L_ / 32, L_ / 128, B_), 128, 0, stream>>>(ctx, mn, W);

    // 2) row / column stats + top-k hit lists (single row pass, two col passes)
    s2sm_row_stats<<<dim3(L_, B_), 256, 0, stream>>>(
        W, rowKth, rowInv, rowStd, rowHitCnt, rowHitIdx, rowHitVal);
    s2sm_col_stats<<<dim3(L_ / 256, B_), 256, 0, stream>>>(
        W, colKth, colInv, colStd, colHitCnt, colHitIdx, colHitVal);

    // 3) sparse attention applies (no further W traffic)
    s2sm_scatter_c<<<dim3(L_, B_), 128, 0, stream>>>(
        ctx, rowInv, rowHitCnt, rowHitIdx, rowHitVal, attC);
    s2sm_gather_m_outc<<<dim3(L_, B_), 128, 0, stream>>>(
        mn, ctx, colInv, colHitCnt, colHitIdx, colHitVal, rowStd, outc);
    s2sm_outm<<<8192, 256, 0, stream>>>(mn, attC, colStd, outm);

    // 4) conv + relu + maxpool: pool_c at offset 0, pool_m at offset F
    const int convBlocks = (L_ - KS_ + 1 + CONV_TL - 1) / CONV_TL;
    s2sm_conv_pool<<<dim3(convBlocks, B_), 128, 0, stream>>>(outc, conv_w, conv_b, out, 0);
    s2sm_conv_pool<<<dim3(convBlocks, B_), 128, 0, stream>>>(outm, conv_w, conv_b, out, F_);
}